// Capsule_52690658787426
// MI455X (gfx1250) — compile-verified
//
#include <hip/hip_runtime.h>
#include <hip/hip_bf16.h>
#include <math.h>

// Problem constants (from reference)
#define BATCH     64
#define INPUT_DIM 512
#define SEQ       2048
#define NCAP      10
#define DCAP      16
#define OUT_CH    (NCAP * DCAP)   // 160
#define KSTEPS    (INPUT_DIM / 4) // 128 WMMA K-steps

typedef __attribute__((ext_vector_type(2))) float v2f;
typedef __attribute__((ext_vector_type(8))) float v8f;

// ---------------------------------------------------------------------------
// Kernel 0: pack W [160][512] into fragment-major layout
//   Wf[ks][nt][lane] = { W[nt*16+n][ks*4+ksel], W[nt*16+n][ks*4+ksel+1] }
// with n = lane&15, ksel = (lane>>4)*2 — i.e. exactly the per-lane v2f the
// V_WMMA_F32_16X16X4_F32 B-operand wants. Each B fragment then becomes ONE
// coalesced global_load_b64 (256 B per wave), L2-resident (320 KB).
// ---------------------------------------------------------------------------
__global__ __launch_bounds__(256)
void pack_w(const float* __restrict__ W, float* __restrict__ Wf) {
    int idx = blockIdx.x * 256 + threadIdx.x;     // over 128*10*32 = 40960
    if (idx >= KSTEPS * NCAP * 32) return;
    const int ks   = idx / (NCAP * 32);
    const int rem  = idx - ks * (NCAP * 32);
    const int nt   = rem >> 5;
    const int lane = rem & 31;
    const int n    = lane & 15;
    const int ksel = (lane >> 4) << 1;
    // Two consecutive i-values of one output row: contiguous & 8B-aligned.
    const float2 w = *(const float2*)(W + (size_t)(nt * 16 + n) * INPUT_DIM
                                        + ks * 4 + ksel);
    *(float2*)(Wf + (size_t)idx * 2) = w;
}

// ---------------------------------------------------------------------------
// Kernel 1: u_hat[b][n][l][d] = sum_i u[b][i][l] * W[n*16+d][i]
// Batched GEMM via V_WMMA_F32_16X16X4_F32 (full f32, matching the reference).
// Each wave: 2 M-tiles (32 rows of l) x all 10 N-tiles (= capsules).
// A loads are NON-TEMPORAL: u_vecs (256 MB) is streamed exactly once, and we
// want u_hat (84 MB) + Wf to stay resident in the 192 MB L2 for routing.
// ---------------------------------------------------------------------------
__global__ __launch_bounds__(256)
void gemm_uhat(const float* __restrict__ u, const float* __restrict__ Wf,
               float* __restrict__ uhat) {
    const int lane = threadIdx.x & 31;
    const int wave = threadIdx.x >> 5;
    const int wid  = blockIdx.x * 8 + wave;      // 0..4095
    const int b    = wid >> 6;                    // batch 0..63
    const int tp   = wid & 63;                    // M tile-pair 0..63
    const int l0   = tp * 32;                     // rows [l0, l0+32)

    const int n    = lane & 15;                   // frag row (A) / col (B)
    const int ksel = (lane >> 4) << 1;            // 0 or 2

    const float* Abase = u + (size_t)b * INPUT_DIM * SEQ + l0 + n;  // u[b][.][l0+n]
    const v2f*   Bbase = (const v2f*)Wf + lane;                     // frag-major

    v8f acc0[NCAP];
    v8f acc1[NCAP];
#pragma unroll
    for (int nt = 0; nt < NCAP; ++nt) {
        acc0[nt] = (v8f){0.f, 0.f, 0.f, 0.f, 0.f, 0.f, 0.f, 0.f};
        acc1[nt] = (v8f){0.f, 0.f, 0.f, 0.f, 0.f, 0.f, 0.f, 0.f};
    }

    for (int k = 0; k < KSTEPS; ++k) {
        const float* ap = Abase + (size_t)(k * 4 + ksel) * SEQ;
        v2f a0, a1;
        a0.x = __builtin_nontemporal_load(ap);             // A(m=n, K=ksel)
        a0.y = __builtin_nontemporal_load(ap + SEQ);       // A(m=n, K=ksel+1)
        a1.x = __builtin_nontemporal_load(ap + 16);        // tile 1 rows +16
        a1.y = __builtin_nontemporal_load(ap + SEQ + 16);

        const v2f* bp = Bbase + (size_t)k * (NCAP * 32);
#pragma unroll
        for (int nt = 0; nt < NCAP; ++nt) {
            const v2f bf = bp[nt * 32];                    // one b64, coalesced
            acc0[nt] = __builtin_amdgcn_wmma_f32_16x16x4_f32(
                false, a0, false, bf, (short)0, acc0[nt], false, false);
            acc1[nt] = __builtin_amdgcn_wmma_f32_16x16x4_f32(
                false, a1, false, bf, (short)0, acc1[nt], false, false);
        }
    }

    // Store: D element (M=r+mbase, N=lane&15) -> uhat[(b*10+nt)][l0+M][N]
    // Default RT stores: u_hat must stay L2-resident for the routing passes.
    const int mbase = (lane >> 4) * 8;   // 0 or 8
#pragma unroll
    for (int nt = 0; nt < NCAP; ++nt) {
        float* op = uhat + ((size_t)(b * NCAP + nt) * SEQ + l0 + mbase) * DCAP + n;
#pragma unroll
        for (int r = 0; r < 8; ++r) {
            op[(size_t)r * DCAP]        = acc0[nt][r];     // rows l0+mbase+r
            op[(size_t)(16 + r) * DCAP] = acc1[nt][r];     // rows l0+16+mbase+r
        }
    }
}

// ---------------------------------------------------------------------------
// Kernel 2: c = softmax(b_logits, axis=0) i.e. over the BATCH axis, per (n,l).
// Iteration 0: logits are all zero -> c = 1/64 (no init pass over b needed).
// ---------------------------------------------------------------------------
__global__ __launch_bounds__(256)
void softmax_batch(const float* __restrict__ blog, float* __restrict__ cws, int first) {
    const int idx = blockIdx.x * 256 + threadIdx.x;       // (n,l) column 0..20479
    const int NCL = NCAP * SEQ;
    if (idx >= NCL) return;
    if (first) {
        const float v = 1.0f / (float)BATCH;
        for (int b = 0; b < BATCH; ++b) cws[(size_t)b * NCL + idx] = v;
        return;
    }
    float mx = -INFINITY;
    for (int b = 0; b < BATCH; ++b)
        mx = fmaxf(mx, blog[(size_t)b * NCL + idx]);
    float s = 0.f;
    for (int b = 0; b < BATCH; ++b) {
        float e = __expf(blog[(size_t)b * NCL + idx] - mx);
        cws[(size_t)b * NCL + idx] = e;
        s += e;
    }
    const float inv = 1.0f / s;
    for (int b = 0; b < BATCH; ++b)
        cws[(size_t)b * NCL + idx] *= inv;
}

// ---------------------------------------------------------------------------
// Kernel 3: s[b,n,:] = sum_l c[b,n,l] * u_hat[b,n,l,:]; out = squash(s).
// One workgroup per (b,n); u_hat is L2-resident (84 MB < 192 MB L2).
// ---------------------------------------------------------------------------
__global__ __launch_bounds__(256)
void sum_squash(const float* __restrict__ uhat, const float* __restrict__ cws,
                float* __restrict__ outv) {
    __shared__ float red[256 * DCAP];   // 16 KB
    const int bn  = blockIdx.x;          // b*10+n, 0..639
    const int tid = threadIdx.x;

    const float* up = uhat + (size_t)bn * SEQ * DCAP;
    const float* cp = cws + (size_t)bn * SEQ;

    float acc[DCAP];
#pragma unroll
    for (int d = 0; d < DCAP; ++d) acc[d] = 0.f;

    for (int l = tid; l < SEQ; l += 256) {
        const float c = cp[l];
        const float4* u4 = (const float4*)(up + (size_t)l * DCAP);
#pragma unroll
        for (int q = 0; q < 4; ++q) {
            float4 v = u4[q];
            acc[q * 4 + 0] += c * v.x;
            acc[q * 4 + 1] += c * v.y;
            acc[q * 4 + 2] += c * v.z;
            acc[q * 4 + 3] += c * v.w;
        }
    }
#pragma unroll
    for (int d = 0; d < DCAP; ++d) red[tid * DCAP + d] = acc[d];
    __syncthreads();
    for (int s = 128; s > 0; s >>= 1) {
        if (tid < s) {
#pragma unroll
            for (int d = 0; d < DCAP; ++d)
                red[tid * DCAP + d] += red[(tid + s) * DCAP + d];
        }
        __syncthreads();
    }
    if (tid < DCAP) {
        float sq = 0.f;
#pragma unroll
        for (int d = 0; d < DCAP; ++d) { float x = red[d]; sq += x * x; }
        outv[(size_t)bn * DCAP + tid] = red[tid] * rsqrtf(sq + 1e-10f);
    }
}

// ---------------------------------------------------------------------------
// Kernel 4: b[b,n,l] = sum_d out[b,n,d] * u_hat[b,n,l,d]   (assignment, per ref)
// ---------------------------------------------------------------------------
__global__ __launch_bounds__(256)
void update_logits(const float* __restrict__ uhat, const float* __restrict__ outv,
                   float* __restrict__ blog) {
    const size_t idx = (size_t)blockIdx.x * 256 + threadIdx.x;  // over B*NC*L
    if (idx >= (size_t)BATCH * NCAP * SEQ) return;
    const size_t bn = idx / SEQ;
    const float4* u4 = (const float4*)(uhat + idx * DCAP);
    const float4* o4 = (const float4*)(outv + bn * DCAP);
    float s = 0.f;
#pragma unroll
    for (int q = 0; q < 4; ++q) {
        float4 uu = u4[q];
        float4 oo = o4[q];
        s += uu.x * oo.x + uu.y * oo.y + uu.z * oo.z + uu.w * oo.w;
    }
    blog[idx] = s;
}

// ---------------------------------------------------------------------------
extern "C" void kernel_launch(void* const* d_in, const int* in_sizes, int n_in,
                              void* d_out, int out_size, void* d_ws, size_t ws_size,
                              hipStream_t stream) {
    const float* u = (const float*)d_in[0];   // [64, 512, 2048]
    const float* W = (const float*)d_in[1];   // [160, 512, 1]
    float* out = (float*)d_out;               // [64, 10, 16]

    char* ws = (char*)d_ws;
    float* Wf   = (float*)ws;                                        // 512*160 f32 frag-major
    ws += (size_t)INPUT_DIM * OUT_CH * sizeof(float);
    float* uhat = (float*)ws;                                        // [64][10][2048][16]
    ws += (size_t)BATCH * NCAP * SEQ * DCAP * sizeof(float);
    float* blog = (float*)ws;                                        // [64][10][2048]
    ws += (size_t)BATCH * NCAP * SEQ * sizeof(float);
    float* cws  = (float*)ws;                                        // [64][10][2048]
    ws += (size_t)BATCH * NCAP * SEQ * sizeof(float);
    float* otmp = (float*)ws;                                        // [64][10][16]

    pack_w<<<(KSTEPS * NCAP * 32 + 255) / 256, 256, 0, stream>>>(W, Wf);
    gemm_uhat<<<512, 256, 0, stream>>>(u, Wf, uhat);   // 4096 waves, 8 per block

    const int ncl_blocks = (NCAP * SEQ + 255) / 256;                        // 80
    const int bnl_blocks = (int)(((size_t)BATCH * NCAP * SEQ + 255) / 256); // 5120
    for (int it = 0; it < 3; ++it) {
        softmax_batch<<<ncl_blocks, 256, 0, stream>>>(blog, cws, it == 0 ? 1 : 0);
        float* dst = (it == 2) ? out : otmp;
        sum_squash<<<BATCH * NCAP, 256, 0, stream>>>(uhat, cws, dst);
        if (it < 2)
            update_logits<<<bnl_blocks, 256, 0, stream>>>(uhat, dst, blog);
    }
}